// AutoAggregation_43585328120069
// MI455X (gfx1250) — compile-verified
//
#include <hip/hip_runtime.h>
#include <math.h>

typedef __attribute__((ext_vector_type(2))) float v2f;
typedef __attribute__((ext_vector_type(8))) float v8f;

#define Bb 8
#define Hh 8
#define Ll 1024
#define Ee 512
#define TILE 16
#define QSTRIDE 516   // padded LDS row stride (floats) for q/k/v/res (16B aligned, conflict-free)
#define CSTRIDE 513   // padded LDS row stride for corr (column access conflict-free)
#define TOPK 6

#define OFF_Q    0
#define OFF_K    (OFF_Q + TILE*QSTRIDE)
#define OFF_RE   (OFF_K + TILE*QSTRIDE)
#define OFF_IMN  (OFF_RE + TILE*QSTRIDE)
#define OFF_CORR (OFF_IMN + TILE*QSTRIDE)
#define SMEM_FLOATS (OFF_CORR + TILE*CSTRIDE)

// DFT twiddle tables: COS[a*512+b] = cos(2*pi*a*b/512), SIN likewise (symmetric).
// Exact integer reduction (a*b mod 512) keeps fp32 accuracy high.
__global__ void build_tables(float* __restrict__ cosT, float* __restrict__ sinT) {
    int id = blockIdx.x * blockDim.x + threadIdx.x;   // 0 .. 512*512-1
    int a = id >> 9, b = id & 511;
    int prod = (a * b) & 511;
    float ang = (float)prod * 0.012271846303085129f;  // 2*pi/512
    float s, c;
    __sincosf(ang, &s, &c);
    cosT[id] = c;
    sinT[id] = s;
}

// One workgroup (8 waves) per 16-row tile of the flattened (B*H*L) rows.
// corr = irfft(fft(q)*conj(fft(k))) realized as 6 fp32 WMMA GEMMs against
// shared DFT matrices, then fused top-6 + softmax + delay aggregation.
__launch_bounds__(256)
__global__ void autocorr_fused(const float* __restrict__ q,
                               const float* __restrict__ k,
                               const float* __restrict__ v,
                               const float* __restrict__ cosT,
                               const float* __restrict__ sinT,
                               float* __restrict__ outV,
                               float* __restrict__ outCorrT)
{
    extern __shared__ float smem[];
    float* lq    = smem + OFF_Q;
    float* lk    = smem + OFF_K;
    float* lre   = smem + OFF_RE;
    float* limn  = smem + OFF_IMN;
    float* lcorr = smem + OFF_CORR;
    float* lv    = lq;               // values tile aliases q tile (q dead after forward)

    const int tid  = threadIdx.x;
    const int wave = tid >> 5;
    const int lane = tid & 31;
    const int R0   = blockIdx.x * TILE;      // first flattened row of tile
    const int bb   = R0 >> 13;               // / (H*L)
    const int hh   = (R0 >> 10) & 7;
    const int l0   = R0 & 1023;

    // ---- Phase 0: stage q,k tiles (16x512 each) into LDS ----
    {
        const float4* gq = (const float4*)(q + (size_t)R0 * Ee);
        const float4* gk = (const float4*)(k + (size_t)R0 * Ee);
        #pragma unroll
        for (int j = 0; j < 8; ++j) {
            int i4  = tid + 256 * j;         // 2048 float4 per matrix
            int row = i4 >> 7;               // / (512/4)
            int col = (i4 & 127) << 2;
            float4 a = gq[i4];
            float4 b = gk[i4];
            *(float4*)(lq + row * QSTRIDE + col) = a;
            *(float4*)(lk + row * QSTRIDE + col) = b;
        }
    }
    __syncthreads();

    // WMMA fragment lane mapping (ISA 7.12.2):
    // A 16x4 f32: lanes 0-15 -> M=lane, K = k0+{0,1}; lanes 16-31 -> M=lane-16, K = k0+{2,3}
    // B 4x16 f32: VGPR0 lanes0-15 K=k0 / lanes16-31 K=k0+1; VGPR1 K=k0+2 / k0+3; N = lane&15
    // C/D 16x16:  VGPR r: lanes0-15 M=r, lanes16-31 M=r+8; N = lane&15
    const int mrow = lane & 15;
    const int ksub = (lane >> 4) << 1;   // 0 or 2
    const int bk0  = lane >> 4;          // 0 or 1
    const int ncol = lane & 15;
    const int msel = (lane >> 4) << 3;   // 0 or 8

    // ---- Phase 1: forward DFT (Qc,Qs,Kc,Ks) + pointwise spectrum product ----
    for (int t = 0; t < 4; ++t) {
        const int fbase = ((wave << 2) + t) << 4;      // this wave's 16-freq column tile
        v8f aQc = {}, aQs = {}, aKc = {}, aKs = {};
        const float* pq = lq + mrow * QSTRIDE + ksub;
        const float* pk = lk + mrow * QSTRIDE + ksub;
        const float* pc = cosT + bk0 * Ee + fbase + ncol;
        const float* ps = sinT + bk0 * Ee + fbase + ncol;
        for (int k0 = 0; k0 < Ee; k0 += 4) {
            v2f aq = *(const v2f*)(pq + k0);
            v2f ak = *(const v2f*)(pk + k0);
            v2f bc; bc.x = pc[(size_t)k0 * Ee]; bc.y = pc[(size_t)(k0 + 2) * Ee];
            v2f bs; bs.x = ps[(size_t)k0 * Ee]; bs.y = ps[(size_t)(k0 + 2) * Ee];
            aQc = __builtin_amdgcn_wmma_f32_16x16x4_f32(false, aq, false, bc, (short)0, aQc, false, false);
            aQs = __builtin_amdgcn_wmma_f32_16x16x4_f32(false, aq, false, bs, (short)0, aQs, false, false);
            aKc = __builtin_amdgcn_wmma_f32_16x16x4_f32(false, ak, false, bc, (short)0, aKc, false, false);
            aKs = __builtin_amdgcn_wmma_f32_16x16x4_f32(false, ak, false, bs, (short)0, aKs, false, false);
        }
        // res = q_fft * conj(k_fft); store Re and -Im, pre-scaled by 1/N for the inverse
        const float sc = 1.0f / (float)Ee;
        #pragma unroll
        for (int r = 0; r < 8; ++r) {
            float Rre  = (aQc[r] * aKc[r] + aQs[r] * aKs[r]) * sc;
            float Rimn = (aQs[r] * aKc[r] - aQc[r] * aKs[r]) * sc;
            int m = r + msel;
            lre [m * QSTRIDE + fbase + ncol] = Rre;
            limn[m * QSTRIDE + fbase + ncol] = Rimn;
        }
    }
    __syncthreads();

    // ---- Phase 2: inverse DFT -> corr tile (LDS) + transposed global write ----
    for (int t = 0; t < 4; ++t) {
        const int dbase = ((wave << 2) + t) << 4;      // this wave's 16-delay column tile
        v8f acc = {};
        const float* pre = lre  + mrow * QSTRIDE + ksub;
        const float* pim = limn + mrow * QSTRIDE + ksub;
        const float* pc  = cosT + bk0 * Ee + dbase + ncol;
        const float* ps  = sinT + bk0 * Ee + dbase + ncol;
        for (int f0 = 0; f0 < Ee; f0 += 4) {
            v2f are = *(const v2f*)(pre + f0);
            v2f aim = *(const v2f*)(pim + f0);
            v2f bc; bc.x = pc[(size_t)f0 * Ee]; bc.y = pc[(size_t)(f0 + 2) * Ee];
            v2f bs; bs.x = ps[(size_t)f0 * Ee]; bs.y = ps[(size_t)(f0 + 2) * Ee];
            acc = __builtin_amdgcn_wmma_f32_16x16x4_f32(false, are, false, bc, (short)0, acc, false, false);
            acc = __builtin_amdgcn_wmma_f32_16x16x4_f32(false, aim, false, bs, (short)0, acc, false, false);
        }
        const int d = dbase + ncol;
        #pragma unroll
        for (int r = 0; r < 8; ++r)
            lcorr[(r + msel) * CSTRIDE + d] = acc[r];
        // corr_T layout (B,E,H,L): ((b*512+d)*8+h)*1024 + l ; lane owns 8 consecutive l
        float* gout = outCorrT + ((((size_t)bb * Ee + d) * Hh + hh) << 10) + l0 + msel;
        float4 lo = make_float4(acc[0], acc[1], acc[2], acc[3]);
        float4 hi = make_float4(acc[4], acc[5], acc[6], acc[7]);
        *(float4*)(gout)     = lo;
        *(float4*)(gout + 4) = hi;
    }
    __syncthreads();

    // ---- Phase 3: stage values tile into LDS (aliases q tile) ----
    {
        const float4* gv = (const float4*)(v + (size_t)R0 * Ee);
        #pragma unroll
        for (int j = 0; j < 8; ++j) {
            int i4  = tid + 256 * j;
            int row = i4 >> 7;
            int col = (i4 & 127) << 2;
            float4 a = gv[i4];
            *(float4*)(lv + row * QSTRIDE + col) = a;
        }
    }
    __syncthreads();

    // ---- Phase 4: per-row top-6 (lowest-index tie-break), softmax, delay-gather ----
    for (int rr = 0; rr < 2; ++rr) {
        const int row = (wave << 1) + rr;
        float valv[16];
        const int base = row * CSTRIDE + lane;
        #pragma unroll
        for (int j = 0; j < 16; ++j) valv[j] = lcorr[base + 32 * j];

        float wsel[TOPK]; int dsel[TOPK];
        for (int c = 0; c < TOPK; ++c) {
            float bestv = -INFINITY; int besti = 0x7fffffff;
            #pragma unroll
            for (int j = 0; j < 16; ++j) {
                int idx = lane + 32 * j;
                if (valv[j] > bestv || (valv[j] == bestv && idx < besti)) { bestv = valv[j]; besti = idx; }
            }
            #pragma unroll
            for (int off = 16; off >= 1; off >>= 1) {
                float ov = __shfl_xor(bestv, off, 32);
                int   oi = __shfl_xor(besti, off, 32);
                if (ov > bestv || (ov == bestv && oi < besti)) { bestv = ov; besti = oi; }
            }
            wsel[c] = bestv; dsel[c] = besti;
            #pragma unroll
            for (int j = 0; j < 16; ++j)
                if ((besti & 31) == lane && (besti >> 5) == j) valv[j] = -INFINITY;
        }
        // softmax over 6 selected weights (wsel[0] is the max)
        float ww[TOPK]; float s = 0.f;
        #pragma unroll
        for (int c = 0; c < TOPK; ++c) { ww[c] = expf(wsel[c] - wsel[0]); s += ww[c]; }
        const float inv = 1.f / s;
        const float* vr = lv + row * QSTRIDE;
        float* go = outV + (size_t)(R0 + row) * Ee;
        for (int e = lane; e < Ee; e += 32) {
            float a = 0.f;
            #pragma unroll
            for (int c = 0; c < TOPK; ++c)
                a += ww[c] * vr[(e + dsel[c]) & 511];
            go[e] = a * inv;
        }
    }
}

extern "C" void kernel_launch(void* const* d_in, const int* in_sizes, int n_in,
                              void* d_out, int out_size, void* d_ws, size_t ws_size,
                              hipStream_t stream) {
    (void)in_sizes; (void)n_in; (void)out_size; (void)ws_size;
    const float* q = (const float*)d_in[0];
    const float* k = (const float*)d_in[1];
    const float* v = (const float*)d_in[2];
    float* cosT = (float*)d_ws;                 // 512*512 f32
    float* sinT = cosT + 512 * 512;             // 512*512 f32  (2 MB total in ws)
    float* outV     = (float*)d_out;                          // (B,H,L,E)
    float* outCorrT = outV + (size_t)Bb * Hh * Ll * Ee;       // (B,E,H,L)

    const size_t smem_bytes = (size_t)SMEM_FLOATS * sizeof(float);
    hipFuncSetAttribute((const void*)autocorr_fused,
                        hipFuncAttributeMaxDynamicSharedMemorySize,
                        (int)smem_bytes);

    build_tables<<<(512 * 512) / 256, 256, 0, stream>>>(cosT, sinT);
    autocorr_fused<<<(Bb * Hh * Ll) / TILE, 256, smem_bytes, stream>>>(
        q, k, v, cosT, sinT, outV, outCorrT);
}